// Prop_47923245089055
// MI455X (gfx1250) — compile-verified
//
#include <hip/hip_runtime.h>
#include <hip/hip_bf16.h>

#define B_ 4
#define D_ 81
#define H_ 256
#define W_ 256
#define CHUNK 27          // 81 = 3 * 27 d-lines per TDM tile
#define NCHUNK 3

typedef unsigned int uint32x4 __attribute__((ext_vector_type(4)));
typedef int          int32x8  __attribute__((ext_vector_type(8)));
typedef int          int32x4  __attribute__((ext_vector_type(4)));

// ---------------------------------------------------------------------------
// TDM: load a [CHUNK lines x W_ floats] tile (line stride = H*W elements,
// i.e. consecutive d-slices of one (b,h) row) from global into LDS.
// Descriptor layout per CDNA5 ISA.
// Toolchain here is clang-23 (therock): 6-arg builtin form.
// ---------------------------------------------------------------------------
__device__ __forceinline__ void tdm_load_chunk(const float* gptr, unsigned lds_byte_off) {
    unsigned long long addr = (unsigned long long)(size_t)gptr;
    uint32x4 g0;
    g0[0] = 1u;                                   // count=1 (valid), no gather
    g0[1] = lds_byte_off;                         // lds_addr (bytes)
    g0[2] = (unsigned)(addr & 0xFFFFFFFFu);       // global_addr[31:0]
    g0[3] = (unsigned)((addr >> 32) & 0x01FFFFFFu) | 0x80000000u; // addr[56:32] | type=2

    int32x8 g1;
    g1[0] = (int)(2u << 16);                      // data_size=2 (4 bytes), mask=0
    g1[1] = (int)((unsigned)W_ << 16);            // tensor_dim0[15:0] = 256
    g1[2] = (int)((unsigned)CHUNK << 16);         // tensor_dim0 hi=0 | tensor_dim1[15:0]=27
    g1[3] = (int)((unsigned)W_ << 16);            // tensor_dim1 hi=0 | tile_dim0=256
    g1[4] = (int)CHUNK;                           // tile_dim1=27 | tile_dim2=0
    g1[5] = (int)(H_ * W_);                       // tensor_dim0_stride[31:0] = 65536 elems
    g1[6] = 0;                                    // stride hi | tensor_dim1_stride lo
    g1[7] = 0;

    int32x4 g2 = {0, 0, 0, 0};
    int32x4 g3 = {0, 0, 0, 0};
    int32x8 g4 = {0, 0, 0, 0, 0, 0, 0, 0};        // extra group (clang-23 6-arg form)
    __builtin_amdgcn_tensor_load_to_lds(g0, g1, g2, g3, g4, 0);
}

// ---------------------------------------------------------------------------
// Pass 1: mincost[b,h,w] = min_d cost[b,d,h,w].
// One block per (b,h) row; TDM double-buffers 27-line tiles into LDS.
// ---------------------------------------------------------------------------
__global__ __launch_bounds__(256) void min_reduce_kernel(const float* __restrict__ cost,
                                                         float* __restrict__ mincost) {
    __shared__ float lbuf[2][CHUNK * W_];         // 2 * 27KB
    const int tid = threadIdx.x;                  // = w
    const int row = blockIdx.x;                   // b*H + h
    const int b = row / H_;
    const int h = row - b * H_;

    const float* base = cost + ((size_t)b * D_ * H_ + h) * W_;   // (b, d=0, h, 0)
    const unsigned lds0 = (unsigned)(size_t)&lbuf[0][0];
    const unsigned lds1 = (unsigned)(size_t)&lbuf[1][0];

    if (tid < 32) {
        tdm_load_chunk(base, lds0);
        __builtin_amdgcn_s_wait_tensorcnt(0);
    }
    __syncthreads();

    float m = __builtin_inff();
    for (int c = 0; c < NCHUNK; ++c) {
        const int cur = c & 1;
        if (c + 1 < NCHUNK && tid < 32) {
            // prefetch next tile into the other buffer while we reduce this one
            tdm_load_chunk(base + (size_t)(c + 1) * CHUNK * H_ * W_, cur ? lds0 : lds1);
        }
        const float* buf = &lbuf[cur][0];
        #pragma unroll
        for (int i = 0; i < CHUNK; ++i)
            m = fminf(m, buf[i * W_ + tid]);
        __syncthreads();                          // everyone done reading buf[cur]
        if (c + 1 < NCHUNK) {
            if (tid < 32) __builtin_amdgcn_s_wait_tensorcnt(0);
            __syncthreads();                      // next tile visible to all waves
        }
    }
    mincost[(size_t)row * W_ + tid] = m;
}

// ---------------------------------------------------------------------------
// Wave32 parallel scan of affine maps m -> E*m + A (8 elements per lane).
// Returns the incoming m for this lane's segment via exclusive scan.
// ---------------------------------------------------------------------------
__device__ __forceinline__ float affine_scan_entry(float a[8], float e[8], int lane) {
    float E = 1.0f, A = 0.0f;
    #pragma unroll
    for (int j = 0; j < 8; ++j) { A = fmaf(e[j], A, a[j]); E *= e[j]; }
    #pragma unroll
    for (int off = 1; off < 32; off <<= 1) {
        float Ep = __shfl_up(E, off, 32);
        float Ap = __shfl_up(A, off, 32);
        if (lane >= off) { A = fmaf(E, Ap, A); E *= Ep; }
    }
    float Aexcl = __shfl_up(A, 1, 32);
    return (lane == 0) ? 0.0f : Aexcl;
}

// ---------------------------------------------------------------------------
// Pass 2a: horizontal directions (edge ch 0 fwd, ch 1 bwd). One wave per row.
// S[b,h,w] = msg_fwd + msg_bwd   (writes S)
// ---------------------------------------------------------------------------
__global__ __launch_bounds__(256) void hscan_kernel(const float* __restrict__ mincost,
                                                    const float* __restrict__ edge,
                                                    float* __restrict__ S) {
    __shared__ float acc[8][W_];
    const int wave = threadIdx.x >> 5, lane = threadIdx.x & 31;
    const int row = blockIdx.x * 8 + wave;        // b*H + h
    const int b = row / H_, h = row - b * H_;
    const float* mc = mincost + (size_t)row * W_;
    const float* e0 = edge + ((size_t)(b * 4 + 0) * H_ + h) * W_;
    const float* e1 = edge + ((size_t)(b * 4 + 1) * H_ + h) * W_;
    float* accw = acc[wave];
    const int base = lane * 8;

    {   // forward (hori_pos)
        float a[8], e[8];
        #pragma unroll
        for (int j = 0; j < 8; ++j) { a[j] = mc[base + j]; e[j] = e0[base + j]; }
        float m = affine_scan_entry(a, e, lane);
        #pragma unroll
        for (int j = 0; j < 8; ++j) { float msg = e[j] * m; accw[base + j] = msg; m = a[j] + msg; }
    }
    {   // backward (hori_neg)
        float a[8], e[8];
        #pragma unroll
        for (int j = 0; j < 8; ++j) { int w = W_ - 1 - (base + j); a[j] = mc[w]; e[j] = e1[w]; }
        float m = affine_scan_entry(a, e, lane);
        #pragma unroll
        for (int j = 0; j < 8; ++j) {
            int w = W_ - 1 - (base + j);
            float msg = e[j] * m; accw[w] += msg; m = a[j] + msg;
        }
    }
    #pragma unroll
    for (int j = 0; j < 8; ++j)
        S[(size_t)row * W_ + base + j] = accw[base + j];
}

// ---------------------------------------------------------------------------
// Pass 2b: vertical directions (edge ch 2 fwd, ch 3 bwd). One wave per column.
// S[b,h,w] += msg_up + msg_down  (RMW; runs after hscan on same stream)
// ---------------------------------------------------------------------------
__global__ __launch_bounds__(256) void vscan_kernel(const float* __restrict__ mincost,
                                                    const float* __restrict__ edge,
                                                    float* __restrict__ S) {
    __shared__ float acc[8][H_];
    const int wave = threadIdx.x >> 5, lane = threadIdx.x & 31;
    const int col = blockIdx.x * 8 + wave;        // b*W + w
    const int b = col / W_, w = col - b * W_;
    const float* mc = mincost + (size_t)b * H_ * W_ + w;       // [h*W_]
    const float* e2 = edge + ((size_t)(b * 4 + 2) * H_) * W_ + w;
    const float* e3 = edge + ((size_t)(b * 4 + 3) * H_) * W_ + w;
    float* accw = acc[wave];
    const int base = lane * 8;

    {   // forward (vert_pos, h ascending)
        float a[8], e[8];
        #pragma unroll
        for (int j = 0; j < 8; ++j) { int hh = base + j; a[j] = mc[(size_t)hh * W_]; e[j] = e2[(size_t)hh * W_]; }
        float m = affine_scan_entry(a, e, lane);
        #pragma unroll
        for (int j = 0; j < 8; ++j) { float msg = e[j] * m; accw[base + j] = msg; m = a[j] + msg; }
    }
    {   // backward (vert_neg, h descending)
        float a[8], e[8];
        #pragma unroll
        for (int j = 0; j < 8; ++j) { int hh = H_ - 1 - (base + j); a[j] = mc[(size_t)hh * W_]; e[j] = e3[(size_t)hh * W_]; }
        float m = affine_scan_entry(a, e, lane);
        #pragma unroll
        for (int j = 0; j < 8; ++j) {
            int hh = H_ - 1 - (base + j);
            float msg = e[j] * m; accw[hh] += msg; m = a[j] + msg;
        }
    }
    #pragma unroll
    for (int j = 0; j < 8; ++j) {
        int hh = base + j;
        size_t idx = ((size_t)b * H_ + hh) * W_ + w;
        S[idx] += accw[hh];
    }
}

// ---------------------------------------------------------------------------
// Pass 3: out[b,d,h,w] = 4*cost[b,d,h,w] + S[b,h,w]   (float4 streaming)
// ---------------------------------------------------------------------------
__global__ __launch_bounds__(256) void combine_kernel(const float* __restrict__ cost,
                                                      const float* __restrict__ S,
                                                      float* __restrict__ out) {
    const size_t i4 = ((size_t)blockIdx.x * 256 + threadIdx.x) * 4;
    const int hw = (int)(i4 & (size_t)(H_ * W_ - 1));   // H*W = 65536 = 2^16
    const int bd = (int)(i4 >> 16);
    const int b = bd / D_;
    const float4 c = *(const float4*)(cost + i4);
    const float4 s = *(const float4*)(S + ((size_t)b << 16) + hw);
    float4 o;
    o.x = 4.0f * c.x + s.x;
    o.y = 4.0f * c.y + s.y;
    o.z = 4.0f * c.z + s.z;
    o.w = 4.0f * c.w + s.w;
    *(float4*)(out + i4) = o;
}

extern "C" void kernel_launch(void* const* d_in, const int* in_sizes, int n_in,
                              void* d_out, int out_size, void* d_ws, size_t ws_size,
                              hipStream_t stream) {
    const float* cost = (const float*)d_in[0];   // [B, D, H, W]
    const float* edge = (const float*)d_in[1];   // [B, 4, H, W]
    float* out = (float*)d_out;                  // [B, D, H, W]

    float* mincost = (float*)d_ws;                            // B*H*W floats (1 MB)
    float* S       = mincost + (size_t)B_ * H_ * W_;          // B*H*W floats (1 MB)

    min_reduce_kernel<<<B_ * H_, 256, 0, stream>>>(cost, mincost);
    hscan_kernel<<<(B_ * H_) / 8, 256, 0, stream>>>(mincost, edge, S);
    vscan_kernel<<<(B_ * W_) / 8, 256, 0, stream>>>(mincost, edge, S);

    const size_t total = (size_t)B_ * D_ * H_ * W_;           // 21,233,664
    combine_kernel<<<(unsigned)(total / (4 * 256)), 256, 0, stream>>>(cost, S, out);
}